// GUME_34883724378517
// MI455X (gfx1250) — compile-verified
//
#include <hip/hip_runtime.h>
#include <math.h>

// ---------------------------------------------------------------------------
// GUME multimodal GNN forward, MI455X (gfx1250, wave32, WMMA + async-to-LDS).
// ---------------------------------------------------------------------------

typedef __attribute__((ext_vector_type(2))) float v2f;
typedef __attribute__((ext_vector_type(8))) float v8f;

#define NUSERS 50000
#define NITEMS 30000
#define NNODES (NUSERS + NITEMS)
#define DIM 64

#if defined(__AMDGCN__) && __has_builtin(__builtin_amdgcn_global_load_async_to_lds_b128) && __has_builtin(__builtin_amdgcn_s_wait_asynccnt)
#define GUME_ASYNC_LDS 1
#else
#define GUME_ASYNC_LDS 0
#endif

#if GUME_ASYNC_LDS
// Builtin prototype (from clang diagnostics): arg0 = int4* in global AS (AS1,
// non-const), arg1 = LDS destination, arg2/arg3 = imm offset / cpol.
typedef int gume_v4i __attribute__((vector_size(16)));
typedef __attribute__((address_space(1))) gume_v4i* gume_gptr;
typedef __attribute__((address_space(3))) gume_v4i* gume_lptr;
#define ASYNC_B128(gsrc, ldst) \
    __builtin_amdgcn_global_load_async_to_lds_b128((gume_gptr)(gsrc), (gume_lptr)(ldst), 0, 0)
#endif

// ===========================================================================
// WMMA GEMM:  Y[M,64] = act( X[M,K] @ W[K,64] + bias )
// 256 threads = 8 wave32 waves. Block tile: 32 rows x 64 cols; each wave owns
// one 16x16 tile -> v_wmma_f32_16x16x4_f32 over K. K staged through
// double-buffered LDS in 32-wide chunks via GLOBAL_LOAD_ASYNC_TO_LDS_B128
// (sync fallback if the builtin is unavailable). K % 32 == 0 at all call
// sites (4096, 384, 64). act: 0=none, 1=sigmoid, 2=tanh.
//
// LDS strides keep async B128 destinations 16B-aligned AND bank-clean:
//   Xs stride 36 floats: 36*r mod 64 = distinct multiples of 4 for r=0..15;
//                        half-waves (K vs K+2) land in disjoint mod-4 classes.
//   Ws stride 72 floats: half-waves read banks {8s..8s+15} / {8s+16..8s+31}.
// ===========================================================================
__global__ __launch_bounds__(256)
void gemm_bias_act_wmma(const float* __restrict__ X, const float* __restrict__ W,
                        const float* __restrict__ bias, float* __restrict__ Y,
                        int M, int K, int act)
{
    __shared__ float Xs[2][32][36];
    __shared__ float Ws[2][32][72];

    const int tid  = threadIdx.x;
    const int lane = tid & 31;
    const int wave = tid >> 5;
    const int row_tile = wave >> 2;   // 0..1
    const int col_tile = wave & 3;    // 0..3
    const int m0 = blockIdx.x * 32;

    // ---- hoisted staging addresses; OOB rows clamp to M-1 (their D-rows are
    //      never stored, so the garbage is harmless and the K-loop stays
    //      branch-free) ----
    const int xr = tid >> 3;             // 0..31 (row within tile)
    const int xc = (tid & 7) * 4;        // 0,4,...,28
    int gxr = m0 + xr; if (gxr >= M) gxr = M - 1;
    const float* xsrc  = X + (size_t)gxr * K + xc;
    const int wr = tid >> 4;             // 0..15
    const int wc = (tid & 15) * 4;       // 0..60
    const float* wsrc0 = W + (size_t)wr * 64 + wc;
    const float* wsrc1 = W + (size_t)(wr + 16) * 64 + wc;

    v8f c = {};   // 16x16 f32 accumulator: VGPR i = row i (lanes 0-15) / i+8

    // A-fragment (16x4 f32): lane&15 = M row, half-wave selects K | K+2
    const int kk   = (lane >> 4) * 2;
    const int arow = row_tile * 16 + (lane & 15);
    const int bcol = col_tile * 16 + (lane & 15);

#if GUME_ASYNC_LDS
    // -------- prologue: stage chunk 0 into buffer 0 (ASYNCcnt-tracked) ----
    ASYNC_B128(xsrc,  &Xs[0][xr][xc]);
    ASYNC_B128(wsrc0, &Ws[0][wr][wc]);
    ASYNC_B128(wsrc1, &Ws[0][wr + 16][wc]);

    int buf = 0;
    for (int k0 = 0; k0 < K; k0 += 32, buf ^= 1) {
        __builtin_amdgcn_s_wait_asynccnt(0);
        __syncthreads();   // LDS chunk visible to all waves; previous compute
                           // on the other buffer is complete block-wide.
        if (k0 + 32 < K) { // pipeline: issue next chunk into the other buffer
            const int nb = buf ^ 1;
            ASYNC_B128(xsrc + k0 + 32,                 &Xs[nb][xr][xc]);
            ASYNC_B128(wsrc0 + (size_t)(k0 + 32) * 64, &Ws[nb][wr][wc]);
            ASYNC_B128(wsrc1 + (size_t)(k0 + 32) * 64, &Ws[nb][wr + 16][wc]);
        }
        #pragma unroll
        for (int s = 0; s < 32; s += 4) {
            v2f a, b;
            a.x = Xs[buf][arow][s + kk];
            a.y = Xs[buf][arow][s + kk + 1];
            b.x = Ws[buf][s + kk][bcol];
            b.y = Ws[buf][s + kk + 1][bcol];
            c = __builtin_amdgcn_wmma_f32_16x16x4_f32(
                    false, a, false, b, (short)0, c, false, false);
        }
        // no trailing barrier: next iteration's asynccnt-wait + barrier
        // precede any overwrite of this buffer.
    }
#else
    // -------- synchronous fallback: single-buffer staging ----
    for (int k0 = 0; k0 < K; k0 += 32) {
        const float4 xv = *(const float4*)(xsrc + k0);
        Xs[0][xr][xc+0] = xv.x; Xs[0][xr][xc+1] = xv.y;
        Xs[0][xr][xc+2] = xv.z; Xs[0][xr][xc+3] = xv.w;
        const float4 w0 = *(const float4*)(wsrc0 + (size_t)k0 * 64);
        Ws[0][wr][wc+0] = w0.x; Ws[0][wr][wc+1] = w0.y;
        Ws[0][wr][wc+2] = w0.z; Ws[0][wr][wc+3] = w0.w;
        const float4 w1 = *(const float4*)(wsrc1 + (size_t)k0 * 64);
        Ws[0][wr+16][wc+0] = w1.x; Ws[0][wr+16][wc+1] = w1.y;
        Ws[0][wr+16][wc+2] = w1.z; Ws[0][wr+16][wc+3] = w1.w;
        __syncthreads();
        #pragma unroll
        for (int s = 0; s < 32; s += 4) {
            v2f a, b;
            a.x = Xs[0][arow][s + kk];
            a.y = Xs[0][arow][s + kk + 1];
            b.x = Ws[0][s + kk][bcol];
            b.y = Ws[0][s + kk + 1][bcol];
            c = __builtin_amdgcn_wmma_f32_16x16x4_f32(
                    false, a, false, b, (short)0, c, false, false);
        }
        __syncthreads();
    }
#endif

    // ---- epilogue: bias + activation + store (C layout: VGPR i -> M=i / i+8)
    const int col = col_tile * 16 + (lane & 15);
    const float bv = bias[col];
    #pragma unroll
    for (int i = 0; i < 8; ++i) {
        const int gr = m0 + row_tile * 16 + i + ((lane >> 4) << 3);
        if (gr < M) {
            float v = c[i] + bv;
            if (act == 1)      v = 1.f / (1.f + __expf(-v));
            else if (act == 2) v = tanhf(v);
            Y[(size_t)gr * 64 + col] = v;
        }
    }
}

// ===========================================================================
// COO SpMM: y[row,:] += val * x[col,:]  (D=64). 16 lanes per edge, float4 per
// lane -> global_load_b128 gather + 4x global_atomic_add_f32 scatter (outputs
// <= 20 MB, resident in the 192 MB L2, so atomics resolve on-chip).
// ===========================================================================
__global__ __launch_bounds__(256)
void spmm_coo_atomic(const int* __restrict__ rows, const int* __restrict__ cols,
                     const float* __restrict__ vals, const float* __restrict__ x,
                     float* __restrict__ y, int nnz)
{
    const int e = blockIdx.x * 16 + (threadIdx.x >> 4);
    if (e >= nnz) return;
    const int l = threadIdx.x & 15;
    const int r = rows[e];
    const int cidx = cols[e];
    const float v = vals[e];
    const float4 xv = *(const float4*)(x + (size_t)cidx * 64 + l * 4);
    float* yp = y + (size_t)r * 64 + l * 4;
    atomicAdd(yp + 0, v * xv.x);
    atomicAdd(yp + 1, v * xv.y);
    atomicAdd(yp + 2, v * xv.z);
    atomicAdd(yp + 3, v * xv.w);
}

// ===========================================================================
// Elementwise utilities (float4 granularity; all sizes are multiples of 64)
// ===========================================================================
__global__ void fill_zero4(float4* p, long n4) {
    long i = (long)blockIdx.x * blockDim.x + threadIdx.x;
    if (i < n4) p[i] = make_float4(0.f, 0.f, 0.f, 0.f);
}
__global__ void copy4(float4* __restrict__ d, const float4* __restrict__ s, long n4) {
    long i = (long)blockIdx.x * blockDim.x + threadIdx.x;
    if (i < n4) d[i] = s[i];
}
__global__ void add4(float4* __restrict__ d, const float4* __restrict__ s, long n4) {
    long i = (long)blockIdx.x * blockDim.x + threadIdx.x;
    if (i < n4) {
        float4 a = d[i]; const float4 b = s[i];
        a.x += b.x; a.y += b.y; a.z += b.z; a.w += b.w;
        d[i] = a;
    }
}
__global__ void scale4(float4* p, float s, long n4) {
    long i = (long)blockIdx.x * blockDim.x + threadIdx.x;
    if (i < n4) {
        float4 a = p[i];
        a.x *= s; a.y *= s; a.z *= s; a.w *= s;
        p[i] = a;
    }
}

// att[m] = dot(T[m,:64], w[:64])
__global__ void rowdot64(const float* __restrict__ T, const float* __restrict__ w,
                         float* __restrict__ att, int M)
{
    int r = blockIdx.x * blockDim.x + threadIdx.x;
    if (r >= M) return;
    const float4* t  = (const float4*)(T + (size_t)r * 64);
    const float4* wv = (const float4*)w;
    float s = 0.f;
    #pragma unroll
    for (int i = 0; i < 16; ++i) {
        const float4 a = t[i], b = wv[i];
        s += a.x*b.x + a.y*b.y + a.z*b.z + a.w*b.w;
    }
    att[r] = s;
}

// out = content + (p1*(ie-common) + p2*(te-common) + common)/3,
// common = softmax([ai,at]) . [ie,te]
__global__ void fuse_final(const float* __restrict__ content,
                           const float* __restrict__ ie, const float* __restrict__ te,
                           const float* __restrict__ ai, const float* __restrict__ at,
                           const float* __restrict__ p1, const float* __restrict__ p2,
                           float* __restrict__ out, long n4)
{
    long i = (long)blockIdx.x * blockDim.x + threadIdx.x;
    if (i >= n4) return;
    const long row = i >> 4;   // 16 float4 per 64-wide row
    const float a0 = ai[row], a1 = at[row];
    const float m  = fmaxf(a0, a1);
    float w0 = __expf(a0 - m), w1 = __expf(a1 - m);
    const float inv = 1.f / (w0 + w1);
    w0 *= inv; w1 *= inv;
    const float4 e = ((const float4*)ie)[i];
    const float4 t = ((const float4*)te)[i];
    const float4 cc = ((const float4*)content)[i];
    const float4 g1 = ((const float4*)p1)[i];
    const float4 g2 = ((const float4*)p2)[i];
    float4 r;
    {
        float cm = w0*e.x + w1*t.x;
        r.x = cc.x + (g1.x*(e.x-cm) + g2.x*(t.x-cm) + cm) * (1.f/3.f);
        cm = w0*e.y + w1*t.y;
        r.y = cc.y + (g1.y*(e.y-cm) + g2.y*(t.y-cm) + cm) * (1.f/3.f);
        cm = w0*e.z + w1*t.z;
        r.z = cc.z + (g1.z*(e.z-cm) + g2.z*(t.z-cm) + cm) * (1.f/3.f);
        cm = w0*e.w + w1*t.w;
        r.w = cc.w + (g1.w*(e.w-cm) + g2.w*(t.w-cm) + cm) * (1.f/3.f);
    }
    ((float4*)out)[i] = r;
}

// ===========================================================================
// Host orchestration
// ===========================================================================
static inline void zero_buf(float* p, long n, hipStream_t s) {
    long n4 = n >> 2;
    fill_zero4<<<(unsigned)((n4 + 255) / 256), 256, 0, s>>>((float4*)p, n4);
}
static inline void copy_buf(float* d, const float* src, long n, hipStream_t s) {
    long n4 = n >> 2;
    copy4<<<(unsigned)((n4 + 255) / 256), 256, 0, s>>>((float4*)d, (const float4*)src, n4);
}
static inline void add_buf(float* d, const float* src, long n, hipStream_t s) {
    long n4 = n >> 2;
    add4<<<(unsigned)((n4 + 255) / 256), 256, 0, s>>>((float4*)d, (const float4*)src, n4);
}
static inline void gemm(const float* X, const float* W, const float* b, float* Y,
                        int M, int K, int act, hipStream_t s) {
    gemm_bias_act_wmma<<<(M + 31) / 32, 256, 0, s>>>(X, W, b, Y, M, K, act);
}
static inline void spmm(const int* r, const int* c, const float* v,
                        const float* x, float* y, int nnz, hipStream_t s) {
    spmm_coo_atomic<<<(nnz + 15) / 16, 256, 0, s>>>(r, c, v, x, y, nnz);
}

extern "C" void kernel_launch(void* const* d_in, const int* in_sizes, int n_in,
                              void* d_out, int out_size, void* d_ws, size_t ws_size,
                              hipStream_t stream)
{
    // ---- inputs (setup_inputs order) ----
    const float* user_emb = (const float*)d_in[0];
    const float* item_emb = (const float*)d_in[1];
    const float* ext_img  = (const float*)d_in[2];
    const float* ext_txt  = (const float*)d_in[3];
    const float* v_feat   = (const float*)d_in[4];
    const float* t_feat   = (const float*)d_in[5];
    const float* W_ir = (const float*)d_in[6];  const float* b_ir = (const float*)d_in[7];
    const float* W_it = (const float*)d_in[8];  const float* b_it = (const float*)d_in[9];
    const float* W_tr = (const float*)d_in[10]; const float* b_tr = (const float*)d_in[11];
    const float* W_tt = (const float*)d_in[12]; const float* b_tt = (const float*)d_in[13];
    const float* W_a1 = (const float*)d_in[14]; const float* b_a1 = (const float*)d_in[15];
    const float* w_a2 = (const float*)d_in[16];
    const float* W_ib = (const float*)d_in[17]; const float* b_ib = (const float*)d_in[18];
    const float* W_tb = (const float*)d_in[19]; const float* b_tb = (const float*)d_in[20];
    const float* adj_val = (const float*)d_in[21];
    const float* r_val   = (const float*)d_in[22];
    const float* img_val = (const float*)d_in[23];
    const float* txt_val = (const float*)d_in[24];
    const int* adj_row = (const int*)d_in[25];
    const int* adj_col = (const int*)d_in[26];
    const int* r_row   = (const int*)d_in[27];
    const int* r_col   = (const int*)d_in[28];
    const int* img_row = (const int*)d_in[29];
    const int* img_col = (const int*)d_in[30];
    const int* txt_row = (const int*)d_in[31];
    const int* txt_col = (const int*)d_in[32];
    const int UI_NNZ = in_sizes[21];
    const int R_NNZ  = in_sizes[22];
    const int II_NNZ = in_sizes[23];

    // ---- workspace layout (floats), ~134 MB total ----
    float* ws = (float*)d_ws;
    const long ND = (long)NNODES * DIM;   // 5.12M
    const long ID = (long)NITEMS * DIM;   // 1.92M
    const long UD = (long)NUSERS * DIM;   // 3.20M
    float* ego   = ws;            // [NNODES,64]  (later: txt preference gate P2)
    float* acc   = ego  + ND;     // [NNODES,64]  -> content_embeds
    float* tmp   = acc  + ND;     // [NNODES,64]  spmm scratch / tanh T / gate P1
    float* red   = tmp  + ND;     // [NITEMS,64]  reduce-GEMM intermediate
    float* imf   = red  + ID;     // [NITEMS,64]  image_feats
    float* txf   = imf  + ID;     // [NITEMS,64]  text_feats
    float* iia   = txf  + ID;     // [NITEMS,64]  conv_ii ping buffer
    float* iemb  = iia  + ID;     // [NNODES,64]  image_embeds = [img_user; img_item]
    float* temb  = iemb + ND;     // [NNODES,64]  text_embeds
    float* att_i = temb + ND;     // [NNODES]
    float* att_t = att_i + NNODES;

    float* img_item = iemb + UD;  // rows [NUSERS, NNODES)
    float* txt_item = temb + UD;
    float* out = (float*)d_out;

    // ---- 1) modality space transforms (WMMA GEMMs) ----
    gemm(v_feat, W_ir, b_ir, red, NITEMS, 4096, 0, stream);   // big HBM-bound GEMM
    gemm(red,    W_it, b_it, imf, NITEMS,   64, 1, stream);   // sigmoid
    gemm(t_feat, W_tr, b_tr, red, NITEMS,  384, 0, stream);
    gemm(red,    W_tt, b_tt, txf, NITEMS,   64, 1, stream);   // sigmoid

    // ---- 2) conv_ui: 3-layer propagation over joint graph, mean of layers ----
    copy_buf(ego, user_emb, UD, stream);
    copy_buf(ego + UD, item_emb, ID, stream);
    copy_buf(acc, ego, ND, stream);
    // layer 1: tmp = A @ ego
    zero_buf(tmp, ND, stream);
    spmm(adj_row, adj_col, adj_val, ego, tmp, UI_NNZ, stream);
    add_buf(acc, tmp, ND, stream);
    // layer 2: ego = A @ tmp
    zero_buf(ego, ND, stream);
    spmm(adj_row, adj_col, adj_val, tmp, ego, UI_NNZ, stream);
    add_buf(acc, ego, ND, stream);
    // layer 3: tmp = A @ ego
    zero_buf(tmp, ND, stream);
    spmm(adj_row, adj_col, adj_val, ego, tmp, UI_NNZ, stream);
    add_buf(acc, tmp, ND, stream);
    // content = acc / 4
    scale4<<<(unsigned)((ND/4 + 255)/256), 256, 0, stream>>>((float4*)acc, 0.25f, ND/4);

    // ---- 3) conv_ii (2 layers) + user modality, image then text ----
    zero_buf(iia, ID, stream);
    spmm(img_row, img_col, img_val, imf, iia, II_NNZ, stream);
    zero_buf(img_item, ID, stream);
    spmm(img_row, img_col, img_val, iia, img_item, II_NNZ, stream);
    zero_buf(iemb, UD, stream);
    spmm(r_row, r_col, r_val, img_item, iemb, R_NNZ, stream);
    add_buf(iemb, ext_img, UD, stream);

    zero_buf(iia, ID, stream);
    spmm(txt_row, txt_col, txt_val, txf, iia, II_NNZ, stream);
    zero_buf(txt_item, ID, stream);
    spmm(txt_row, txt_col, txt_val, iia, txt_item, II_NNZ, stream);
    zero_buf(temb, UD, stream);
    spmm(r_row, r_col, r_val, txt_item, temb, R_NNZ, stream);
    add_buf(temb, ext_txt, UD, stream);

    // ---- 4) coarse attention: att = tanh(X@W1+b1) @ w2 ----
    gemm(iemb, W_a1, b_a1, tmp, NNODES, 64, 2, stream);       // tanh
    rowdot64<<<(NNODES + 255)/256, 256, 0, stream>>>(tmp, w_a2, att_i, NNODES);
    gemm(temb, W_a1, b_a1, tmp, NNODES, 64, 2, stream);
    rowdot64<<<(NNODES + 255)/256, 256, 0, stream>>>(tmp, w_a2, att_t, NNODES);

    // ---- 5) behavior-aware preference gates (sigmoid GEMMs) ----
    gemm(acc, W_ib, b_ib, tmp, NNODES, 64, 1, stream);        // P1 (reuses tmp)
    gemm(acc, W_tb, b_tb, ego, NNODES, 64, 1, stream);        // P2 (reuses ego)

    // ---- 6) fused softmax / separation / gating / residual ----
    const long n4 = ND / 4;
    fuse_final<<<(unsigned)((n4 + 255)/256), 256, 0, stream>>>(
        acc, iemb, temb, att_i, att_t, tmp, ego, out, n4);
}